// PFASModule_34059090657735
// MI455X (gfx1250) — compile-verified
//
#include <hip/hip_runtime.h>
#include <hip/hip_bf16.h>
#include <math.h>

// Problem constants (match reference)
#define BATCH   2
#define NPER    8192
#define NTOT    (BATCH * NPER)
#define CIN     32
#define KNN     16
#define NTILES  (NPER / 16)

typedef float v2f __attribute__((ext_vector_type(2)));
typedef float v4f __attribute__((ext_vector_type(4)));
typedef float v8f __attribute__((ext_vector_type(8)));

#define BIGF 1.0e30f

// -----------------------------------------------------------------------------
// Pass 1: per-cloud BatchNorm statistics (sum, sumsq of h = feat@W1^T + b1)
// plus a packed per-point float4 (x, y, z, |c|^2). The packed array makes the
// hot-loop A-fragment load a single uniform global_load_b64 per lane
// (lo lanes read .xy at offset 0, hi lanes read .zw at offset 8).
// Grid: 64 blocks x 256 threads, one thread per point. Blocks never straddle
// clouds (8192 / 256 = 32 blocks per cloud).
// -----------------------------------------------------------------------------
__global__ void __launch_bounds__(256)
pfas_stats_kernel(const float* __restrict__ feat,
                  const float* __restrict__ coord,
                  const float* __restrict__ W1,
                  const float* __restrict__ b1,
                  float* __restrict__ stats,   // [2][64]: sum[32], sumsq[32]
                  v4f*   __restrict__ pack4)   // [NTOT] (x,y,z,|c|^2)
{
    __shared__ float sW1[CIN * CIN];
    __shared__ float ssum[CIN];
    __shared__ float ssq[CIN];

    const int t = threadIdx.x;
    for (int i = t; i < CIN * CIN; i += 256) sW1[i] = W1[i];
    if (t < CIN) { ssum[t] = 0.0f; ssq[t] = 0.0f; }
    __syncthreads();

    const int pt    = blockIdx.x * 256 + t;   // global point id
    const int cloud = pt >> 13;               // pt / 8192

    float f[CIN];
#pragma unroll
    for (int k = 0; k < CIN; ++k) f[k] = feat[pt * CIN + k];

    const float cx = coord[pt * 3 + 0];
    const float cy = coord[pt * 3 + 1];
    const float cz = coord[pt * 3 + 2];
    v4f p; p.x = cx; p.y = cy; p.z = cz; p.w = cx * cx + cy * cy + cz * cz;
    pack4[pt] = p;

#pragma unroll
    for (int j = 0; j < CIN; ++j) {
        float h = b1[j];
#pragma unroll
        for (int k = 0; k < CIN; ++k) h = fmaf(sW1[j * CIN + k], f[k], h);
        atomicAdd(&ssum[j], h);
        atomicAdd(&ssq[j], h * h);
    }
    __syncthreads();

    if (t < CIN) {
        atomicAdd(&stats[cloud * 64 + t],      ssum[t]);
        atomicAdd(&stats[cloud * 64 + 32 + t], ssq[t]);
    }
}

// -----------------------------------------------------------------------------
// Pass 2: main kernel. One block = 128 queries (8 waves x 16 queries).
// Each wave sweeps all 512 candidate tiles of its cloud with
// V_WMMA_F32_16X16X4_F32, producing per-lane (d^2 - |q|^2) values:
//   A (16x4) row m = ( x_m, y_m, z_m, |c_m|^2 )
//   B (4x16) col n = ( -2*qx_n, -2*qy_n, -2*qz_n, 1 )
//   D[m][n] = -2*<c_m,q_n> + |c_m|^2  ->  d2 = D + |q_n|^2
// Fragment layout (ISA 7.12.2, f32):
//   A: lanes 0-15 hold (K0,K1), lanes 16-31 hold (K2,K3), M = lane&15
//   B: lanes 0-15 hold rows (K0,K1), lanes 16-31 rows (K2,K3), N = lane&15
//   D: lane = N (query), VGPR r = M=r (lanes<16) or M=r+8 (lanes>=16)
// A-fragments come from the packed (x,y,z,|c|^2) array: one uniform b64 load
// per lane per tile, double-buffered so the load latency hides under the
// top-K selection work.
// -----------------------------------------------------------------------------
__global__ void __launch_bounds__(256)
pfas_knn_kernel(const float* __restrict__ coord,
                const float* __restrict__ feat,
                const float* __restrict__ W1,
                const float* __restrict__ b1,
                const float* __restrict__ gamma,
                const float* __restrict__ beta,
                const float* __restrict__ W2,
                const float* __restrict__ b2,
                const float* __restrict__ stats,
                const v4f*   __restrict__ pack4,
                float* __restrict__ out)
{
    __shared__ float sW1[CIN * CIN];
    __shared__ float smd[8][32][KNN];
    __shared__ int   smi[8][32][KNN];

    const int t = threadIdx.x;
    for (int i = t; i < CIN * CIN; i += 256) sW1[i] = W1[i];

    const int wave  = t >> 5;
    const int lane  = t & 31;
    const bool hi   = (lane >= 16);
    const int cloud = blockIdx.x >> 6;                       // 64 blocks/cloud
    const int qloc  = (blockIdx.x & 63) * 128 + wave * 16 + (lane & 15);
    const int q     = cloud * NPER + qloc;                   // global query id

    const float qx = coord[q * 3 + 0];
    const float qy = coord[q * 3 + 1];
    const float qz = coord[q * 3 + 2];
    const float sqq = qx * qx + qy * qy + qz * qz;

    // B fragment (loop-invariant): rows (-2qx,-2qy) low lanes, (-2qz,1) high
    v2f bfrag;
    bfrag.x = hi ? (-2.0f * qz) : (-2.0f * qx);
    bfrag.y = hi ? 1.0f         : (-2.0f * qy);

    // per-lane top-16 (squared distances), "replace current max" scheme
    float bd[KNN];
    int   bix[KNN];
#pragma unroll
    for (int k = 0; k < KNN; ++k) { bd[k] = BIGF; bix[k] = -1; }
    float cm  = BIGF;   // current max of list
    int   cmp = 0;      // its position

    const int m      = lane & 15;
    const int cbase0 = cloud * NPER;
    const int mhalf  = hi ? 8 : 0;

    // per-lane uniform source pointer into the packed array:
    // lane reads 8 bytes at (cbase0 + tile*16 + m)*16 + (hi ? 8 : 0)
    const char* abase = (const char*)pack4
                      + ((size_t)(cbase0 + m) * 16) + (hi ? 8 : 0);

    // ---- software-pipelined candidate sweep ----
    v2f afcur = *(const v2f*)abase;

    for (int ct = 0; ct < NTILES; ++ct) {
        // issue next tile's A-fragment load before consuming this tile
        v2f afnext = afcur;
        if (ct + 1 < NTILES) {
            afnext = *(const v2f*)(abase + (size_t)(ct + 1) * 256);
        }
        if (ct + 2 < NTILES) {
            __builtin_prefetch(abase + (size_t)(ct + 2) * 256, 0, 3);
        }

        v8f acc = {0.f, 0.f, 0.f, 0.f, 0.f, 0.f, 0.f, 0.f};
        acc = __builtin_amdgcn_wmma_f32_16x16x4_f32(
                  false, afcur, false, bfrag, (short)0, acc, false, false);

        const int mb = cbase0 + ct * 16 + mhalf;
#pragma unroll
        for (int r = 0; r < 8; ++r) {
            float d2 = acc[r] + sqq;
            const int ci2 = mb + r;
            if (ci2 == q) d2 = BIGF;              // exclude self
            if (d2 < cm) {
#pragma unroll
                for (int k = 0; k < KNN; ++k)
                    if (k == cmp) { bd[k] = d2; bix[k] = ci2; }
                cm = -BIGF;
#pragma unroll
                for (int k = 0; k < KNN; ++k)
                    if (bd[k] > cm) { cm = bd[k]; cmp = k; }
            }
        }

        afcur = afnext;
    }

#pragma unroll
    for (int k = 0; k < KNN; ++k) {
        smd[wave][lane][k] = bd[k];
        smi[wave][lane][k] = bix[k];
    }
    __syncthreads();

    if (lane < 16) {
        // ---- merge the two half-lane lists (32 -> 16 smallest) ----
        unsigned used = 0u;
        float dsum = 0.0f;
        float sx = 0, sy = 0, sz = 0;
        float sxx = 0, sxy = 0, sxz = 0, syy = 0, syz = 0, szz = 0;

        for (int k = 0; k < KNN; ++k) {
            float best = BIGF;
            int   bj   = 0;
            for (int j = 0; j < 32; ++j) {
                if ((used >> j) & 1u) continue;
                const int ln = lane + ((j >> 4) << 4);
                const float v = smd[wave][ln][j & 15];
                if (v < best) { best = v; bj = j; }
            }
            used |= (1u << bj);
            const int ln  = lane + ((bj >> 4) << 4);
            const int idx = smi[wave][ln][bj & 15];
            dsum += sqrtf(fmaxf(best, 0.0f));

            const v4f nb = pack4[idx];            // one b128 gather
            sx += nb.x; sy += nb.y; sz += nb.z;
            sxx += nb.x * nb.x; sxy += nb.x * nb.y; sxz += nb.x * nb.z;
            syy += nb.y * nb.y; syz += nb.y * nb.z; szz += nb.z * nb.z;
        }

        // ---- covariance (unbiased, K-1) ----
        const float KF = (float)KNN, KM1 = (float)(KNN - 1);
        const float mx = sx / KF, my = sy / KF, mz = sz / KF;
        const float a00 = (sxx - KF * mx * mx) / KM1;
        const float a01 = (sxy - KF * mx * my) / KM1;
        const float a02 = (sxz - KF * mx * mz) / KM1;
        const float a11 = (syy - KF * my * my) / KM1;
        const float a12 = (syz - KF * my * mz) / KM1;
        const float a22 = (szz - KF * mz * mz) / KM1;

        // ---- symmetric 3x3 eigenvalues (closed form), descending ----
        const float p1 = a01 * a01 + a02 * a02 + a12 * a12;
        const float qm = (a00 + a11 + a22) * (1.0f / 3.0f);
        float e0, e1, e2;
        if (p1 < 1e-22f) {
            e0 = a00; e1 = a11; e2 = a22;
            if (e0 < e1) { float tt = e0; e0 = e1; e1 = tt; }
            if (e1 < e2) { float tt = e1; e1 = e2; e2 = tt; }
            if (e0 < e1) { float tt = e0; e0 = e1; e1 = tt; }
        } else {
            const float d0 = a00 - qm, d1 = a11 - qm, d2m = a22 - qm;
            const float p2 = d0 * d0 + d1 * d1 + d2m * d2m + 2.0f * p1;
            const float p  = sqrtf(p2 * (1.0f / 6.0f));
            const float ip = 1.0f / p;
            const float b00 = d0 * ip, b11 = d1 * ip, b22 = d2m * ip;
            const float b01 = a01 * ip, b02 = a02 * ip, b12 = a12 * ip;
            float detB = b00 * (b11 * b22 - b12 * b12)
                       - b01 * (b01 * b22 - b12 * b02)
                       + b02 * (b01 * b12 - b11 * b02);
            float r = fminf(fmaxf(detB * 0.5f, -1.0f), 1.0f);
            const float phi = acosf(r) * (1.0f / 3.0f);
            e0 = qm + 2.0f * p * cosf(phi);
            e2 = qm + 2.0f * p * cosf(phi + 2.0943951023931953f); // +2pi/3
            e1 = 3.0f * qm - e0 - e2;
        }
        const float inv = 1.0f / (e0 + e1 + e2 + 1e-6f);
        const float lin = (e0 - (e1 + e2)) * inv;
        const float density = 1.0f / (dsum / KF + 1e-6f);

        // ---- feature judge MLP for this query ----
        float fr[CIN];
#pragma unroll
        for (int k = 0; k < CIN; ++k) fr[k] = feat[q * CIN + k];
        const float invN = 1.0f / (float)NPER;
        float l0 = b2[0], l1 = b2[1], l2 = b2[2];
#pragma unroll
        for (int j = 0; j < CIN; ++j) {
            float hv = b1[j];
#pragma unroll
            for (int k = 0; k < CIN; ++k) hv = fmaf(sW1[j * CIN + k], fr[k], hv);
            const float mu  = stats[cloud * 64 + j] * invN;
            const float var = stats[cloud * 64 + 32 + j] * invN - mu * mu;
            hv = (hv - mu) * (1.0f / sqrtf(var + 1e-5f)) * gamma[j] + beta[j];
            hv = fmaxf(hv, 0.0f);
            l0 = fmaf(W2[0 * CIN + j], hv, l0);
            l1 = fmaf(W2[1 * CIN + j], hv, l1);
            l2 = fmaf(W2[2 * CIN + j], hv, l2);
        }
        const float mmax = fmaxf(l0, fmaxf(l1, l2));
        const float ex0 = expf(l0 - mmax);
        const float ex1 = expf(l1 - mmax);
        const float ex2 = expf(l2 - mmax);
        const float isum = 1.0f / (ex0 + ex1 + ex2);
        const float p0v = ex0 * isum, p1v = ex1 * isum, p2v = ex2 * isum;

        const float tower = (density * 2.0f + p0v) * (1.0f / 3.0f);
        const float bg    = (fmaxf(1.0f - lin, 1.0f - density) + p1v) * (1.0f / 3.0f);
        const float line  = (lin * 2.0f + p2v) * (1.0f / 3.0f);

        // grids: TOWER=(.05,.05,.05)  BG=(.4,.4,.4)  LINE=(.1,.1,1.0)
        out[q * 3 + 0] = tower * 0.05f + bg * 0.4f + line * 0.1f + 1e-6f;
        out[q * 3 + 1] = tower * 0.05f + bg * 0.4f + line * 0.1f + 1e-6f;
        out[q * 3 + 2] = tower * 0.05f + bg * 0.4f + line * 1.0f + 1e-6f;
    }
}

// -----------------------------------------------------------------------------
extern "C" void kernel_launch(void* const* d_in, const int* in_sizes, int n_in,
                              void* d_out, int out_size, void* d_ws, size_t ws_size,
                              hipStream_t stream)
{
    const float* feat  = (const float*)d_in[0];
    const float* coord = (const float*)d_in[1];
    const float* W1    = (const float*)d_in[2];
    const float* b1    = (const float*)d_in[3];
    const float* gamma = (const float*)d_in[4];
    const float* beta  = (const float*)d_in[5];
    const float* W2    = (const float*)d_in[6];
    const float* b2    = (const float*)d_in[7];
    // d_in[8] = batch ids: unused (sorted equal-size partition, as in reference)
    float* out = (float*)d_out;

    float* stats = (float*)d_ws;                 // 2*64 floats
    v4f*   pack4 = (v4f*)((float*)d_ws + 128);   // NTOT float4 (16B aligned)

    hipMemsetAsync(d_ws, 0, 128 * sizeof(float), stream);

    pfas_stats_kernel<<<NTOT / 256, 256, 0, stream>>>(feat, coord, W1, b1, stats, pack4);
    pfas_knn_kernel<<<BATCH * (NPER / 128), 256, 0, stream>>>(
        coord, feat, W1, b1, gamma, beta, W2, b2, stats, pack4, out);
}